// DecodeDetectionsFast_68848325755141
// MI455X (gfx1250) — compile-verified
//
#include <hip/hip_runtime.h>
#include <stdint.h>

// ---- problem constants (from reference) ----
#define CLASS_NUM 20
#define N_BOXES   8732
#define LAST      65          // 3*20+5
#define NSORT     16384       // next pow2 >= N_BOXES
#define TOPK      200
#define CONF_T    0.01f
#define IOU_T     0.45f
#define IMG_CLIP  299.0f      // IMG_W - 1

// Generic -> LDS(addrspace 3) offset for async-DMA destination.
__device__ __forceinline__ unsigned lds_off(const void* p) {
  return (unsigned)(uintptr_t)(const __attribute__((address_space(3))) void*)p;
}

// CDNA5 async global->LDS copy (per-lane b32), tracked by ASYNCcnt.
__device__ __forceinline__ void async_ld_b32(unsigned ldsByteOff, const float* g) {
  asm volatile("global_load_async_to_lds_b32 %0, %1, off"
               :: "v"(ldsByteOff), "v"((unsigned long long)(uintptr_t)g)
               : "memory");
}
#define WAIT_ASYNC_0() asm volatile("s_wait_asynccnt 0" ::: "memory")
#define WAIT_ASYNC_2() asm volatile("s_wait_asynccnt 2" ::: "memory")

extern "C" __global__ __launch_bounds__(1024, 1)
void decode_nms_kernel(const float* __restrict__ y, float* __restrict__ out) {
  // ~154 KB LDS total: legal on CDNA5 (320 KB / workgroup), not on CDNA4.
  __shared__ unsigned long long keys[NSORT];       // 128 KB: (conf_bits<<32)|(~idx)
  __shared__ float stage[32][2][44];               // per-wave double buffer (41 used)
  __shared__ unsigned char cls[N_BOXES + 4];       // class id (1..20) per box
  __shared__ float sx1[TOPK], sy1[TOPK], sx2[TOPK], sy2[TOPK];

  const int tid  = threadIdx.x;
  const int b    = blockIdx.x;
  const int lane = tid & 31;
  const int w    = tid >> 5;                       // wave id 0..31
  const float* __restrict__ yb = y + (size_t)b * N_BOXES * LAST;
  float* __restrict__ ob = out + (size_t)b * TOPK * 6;

  // ---- zero this item's output (harness poisons d_out) ----
  for (int i = tid; i < TOPK * 6; i += 1024) ob[i] = 0.0f;

  // ---- pad sort keys beyond N_BOXES with -inf-equivalent (conf bits = 0) ----
  for (int i = N_BOXES + tid; i < NSORT; i += 1024) keys[i] = 0ull;

  // =====================================================================
  // Phase A: decode. One wave per box; lanes 0..19 hold the 20 classes.
  // Prob inputs (cols 20..60, 41 dwords) are staged global->LDS with the
  // CDNA5 async DMA path, double-buffered so the DMA of box i+32 overlaps
  // the reduction of box i.
  // =====================================================================
  {
    // prologue: issue DMA for this wave's first box into buffer 0
    {
      const float* g = yb + (size_t)w * LAST + 20;
      async_ld_b32(lds_off(&stage[w][0][lane]), g + lane);
      if (lane < 9)
        async_ld_b32(lds_off(&stage[w][0][32 + lane]), g + 32 + lane);
    }
    int buf = 0;
    for (int i = w; i < N_BOXES; i += 32) {
      const int inext = i + 32;
      if (inext < N_BOXES) {
        const float* g = yb + (size_t)inext * LAST + 20;
        async_ld_b32(lds_off(&stage[w][buf ^ 1][lane]), g + lane);
        if (lane < 9)
          async_ld_b32(lds_off(&stage[w][buf ^ 1][32 + lane]), g + 32 + lane);
        WAIT_ASYNC_2();          // current buffer's 2 older DMAs are done
      } else {
        WAIT_ASYNC_0();          // drain before leaving the loop
      }
      const float* s = stage[w][buf];
      // probs[c] = y[...,20+c] * y[...,41+c]  (stage index 21+c == col 41+c)
      unsigned long long kk = 0ull;
      if (lane < CLASS_NUM) {
        float p = s[lane] * s[21 + lane];          // p >= 0: bit order == value order
        kk = (((unsigned long long)__float_as_uint(p)) << 6) | (unsigned)(63 - lane);
      }
      // wave32 max-reduce; ties -> smallest class index (matches jnp.argmax)
      for (int off = 16; off > 0; off >>= 1) {
        unsigned long long o = __shfl_xor(kk, off, 32);
        if (o > kk) kk = o;
      }
      if (lane == 0) {
        unsigned cbits = (unsigned)(kk >> 6);
        unsigned bc    = 63u - (unsigned)(kk & 63ull);
        cls[i]  = (unsigned char)(bc + 1u);
        // descending sort key: conf desc, then original index asc (stable argsort)
        keys[i] = (((unsigned long long)cbits) << 32) | (0xFFFFFFFFu - (unsigned)i);
      }
      buf ^= 1;
    }
  }

  // =====================================================================
  // Phase B: in-LDS bitonic sort of 16384 u64 keys, descending.
  // Pair-indexed: each pass touches exactly NSORT/2 = 8192 pairs, so every
  // thread does 8 compare-exchanges per pass with no discarded work.
  // =====================================================================
  for (unsigned k = 2; k <= NSORT; k <<= 1) {
    for (unsigned j = k >> 1; j > 0; j >>= 1) {
      __syncthreads();
      const unsigned jm1 = j - 1;
      for (unsigned p = tid; p < (NSORT / 2); p += 1024) {
        unsigned i  = ((p & ~jm1) << 1) | (p & jm1);
        unsigned i2 = i + j;
        unsigned long long a = keys[i];
        unsigned long long c2 = keys[i2];
        bool desc = ((i & k) == 0);
        if (desc ? (a < c2) : (a > c2)) { keys[i] = c2; keys[i2] = a; }
      }
    }
  }
  __syncthreads();

  // =====================================================================
  // Phase C: serial greedy NMS by wave 0. Candidate order = sorted order.
  // Output rows are the first 200 selections, so exit at count==200
  // (selection can't change earlier rows). Lanes parallelize the IOU max
  // against the selected set; candidate coords come from L2 (input is
  // L2-resident: 145 MB < 192 MB). Next candidate's coords are prefetched
  // (global_prefetch_b8) to hide the serial gather latency.
  // =====================================================================
  if (w == 0) {
    int count = 0;
    for (int s = 0; s < NSORT && count < TOPK; ++s) {
      unsigned long long kk = keys[s];
      // prefetch next candidate's coordinate dwords while we work on this one
      if (s + 1 < NSORT) {
        unsigned long long kn = keys[s + 1];
        unsigned idn = 0xFFFFFFFFu - (unsigned)(kn & 0xFFFFFFFFull);
        if (idn < N_BOXES)
          __builtin_prefetch(yb + (size_t)idn * LAST + 61, 0, 0);
      }
      float conf = __uint_as_float((unsigned)(kk >> 32));
      if (!(conf > CONF_T)) break;                 // sorted: rest all invalid
      unsigned idx = 0xFFFFFFFFu - (unsigned)(kk & 0xFFFFFFFFull);
      const float* g = yb + (size_t)idx * LAST + 61;
      float ax1 = fminf(fmaxf(g[0], 0.f), IMG_CLIP);
      float ay1 = fminf(fmaxf(g[1], 0.f), IMG_CLIP);
      float ax2 = fminf(fmaxf(g[2], 0.f), IMG_CLIP);
      float ay2 = fminf(fmaxf(g[3], 0.f), IMG_CLIP);
      float areaA = fmaxf(ax2 - ax1, 0.f) * fmaxf(ay2 - ay1, 0.f);
      float mx = 0.0f;
      for (int t = lane; t < count; t += 32) {
        float bx1 = sx1[t], by1 = sy1[t], bx2 = sx2[t], by2 = sy2[t];
        float ix1 = fmaxf(ax1, bx1), iy1 = fmaxf(ay1, by1);
        float ix2 = fminf(ax2, bx2), iy2 = fminf(ay2, by2);
        float inter = fmaxf(ix2 - ix1, 0.f) * fmaxf(iy2 - iy1, 0.f);
        float areaB = fmaxf(bx2 - bx1, 0.f) * fmaxf(by2 - by1, 0.f);
        float uni = areaA + areaB - inter;
        float iou = (uni > 0.f) ? (inter / fmaxf(uni, 1e-8f)) : 0.f;
        mx = fmaxf(mx, iou);
      }
      for (int off = 16; off > 0; off >>= 1)
        mx = fmaxf(mx, __shfl_xor(mx, off, 32));
      if (mx <= IOU_T) {
        if (lane == 0) {
          sx1[count] = ax1; sy1[count] = ay1; sx2[count] = ax2; sy2[count] = ay2;
          float* o = ob + (size_t)count * 6;
          o[0] = (float)cls[idx];
          o[1] = conf;
          o[2] = ax1; o[3] = ay1; o[4] = ax2; o[5] = ay2;
        }
        ++count;
      }
    }
  }
}

extern "C" void kernel_launch(void* const* d_in, const int* in_sizes, int n_in,
                              void* d_out, int out_size, void* d_ws, size_t ws_size,
                              hipStream_t stream) {
  (void)n_in; (void)d_ws; (void)ws_size; (void)out_size;
  const float* y = (const float*)d_in[0];
  float* out = (float*)d_out;
  int batch = in_sizes[0] / (N_BOXES * LAST);      // = 64
  decode_nms_kernel<<<dim3(batch), dim3(1024), 0, stream>>>(y, out);
}